// TransR_48430051229799
// MI455X (gfx1250) — compile-verified
//
#include <hip/hip_runtime.h>

// ---------------------------------------------------------------------------
// TransR (N_MODE==0):  out[b] = (x2[b] @ M_r - x1[b]) @ inv(M_r),
//   M_r = transfer_matrix[rel[b]] as 256x256 fp32.  x0 is unused in mode 0.
//
// One workgroup per relation:
//   - M -> LDS via TDM tensor_load_to_lds (pad 1 DWORD / 256 DWORDs => the
//     stride-257 conflict-free LDS layout), s_wait_tensorcnt to sync
//   - T = X2 @ M - X1 via V_WMMA_F32_16X16X4_F32
//   - blocked (panel=16) in-place Gauss-Jordan inverse: scalar panel factor +
//     rank-16 trailing updates on the fp32 WMMA pipe (~15/16 of the flops)
//   - OUT = T @ Minv via WMMA, scattered through the per-relation permutation
// ---------------------------------------------------------------------------

#define EMB        256
#define RELS       1000
#define BATCH      4096
#define MS_STRIDE  257      // padded LDS row stride for M (gcd(257,64)=1)
#define XS_STRIDE  260      // padded LDS row stride for 16-row staging (16B rows)

typedef __attribute__((ext_vector_type(2))) float        v2f;
typedef __attribute__((ext_vector_type(8))) float        v8f;
typedef __attribute__((ext_vector_type(4))) unsigned int v4u;
typedef __attribute__((ext_vector_type(8))) int          v8i;
typedef __attribute__((ext_vector_type(4))) int          v4i;

// ---- workspace layout (in 4-byte units) -----------------------------------
#define WS_COUNTS   0       // 1000
#define WS_OFFSETS  1024    // 1001
#define WS_CURSOR   2112    // 1000
#define WS_PERM     3200    // 4096
#define WS_T        8192    // 4096*256 floats (4 MB)

__global__ void k_zero(int* counts) {
    int i = blockIdx.x * blockDim.x + threadIdx.x;
    if (i < RELS) counts[i] = 0;
}

__global__ void k_count(const int* __restrict__ rel, int* __restrict__ counts) {
    int b = blockIdx.x * blockDim.x + threadIdx.x;
    if (b < BATCH) atomicAdd(&counts[rel[b]], 1);
}

__global__ void k_scan(const int* __restrict__ counts,
                       int* __restrict__ offsets,
                       int* __restrict__ cursor) {
    if (blockIdx.x == 0 && threadIdx.x == 0) {
        int acc = 0;
        for (int r = 0; r < RELS; ++r) {
            offsets[r] = acc;
            cursor[r]  = acc;
            acc += counts[r];
        }
        offsets[RELS] = acc;
    }
}

__global__ void k_scatter(const int* __restrict__ rel,
                          int* __restrict__ cursor,
                          int* __restrict__ perm) {
    int b = blockIdx.x * blockDim.x + threadIdx.x;
    if (b < BATCH) {
        int r = rel[b];
        int pos = atomicAdd(&cursor[r], 1);
        perm[pos] = b;
    }
}

// ---------------------------------------------------------------------------
__global__ void k_transr(const float* __restrict__ x1,
                         const float* __restrict__ x2,
                         const float* __restrict__ tm,
                         const int*  __restrict__ offsets,
                         const int*  __restrict__ perm,
                         float* __restrict__ T,
                         float* __restrict__ out) {
    extern __shared__ float lds[];
    float* Ms   = lds;                        // 256 x (stride 257)
    float* Xs   = lds + EMB * MS_STRIDE;      // 16  x (stride 260) staging
    float* fcol = Xs  + 16 * XS_STRIDE;       // 256  pivot column cache
    float* prow = fcol + EMB;                 // 16   scaled pivot row (panel)

    const int r    = blockIdx.x;
    const int tid  = threadIdx.x;
    const int base = offsets[r];
    const int nr   = offsets[r + 1] - base;
    if (nr == 0) return;                      // uniform exit: relation unused

    const int wave = tid >> 5;
    const int lane = tid & 31;
    const int lcol = lane & 15;               // B/C/D column within 16-wide tile
    const int lrowbase = (lane < 16) ? 0 : 8; // C/D: VGPR v -> row v (+8 hi half)
    const int arow = lane & 15;               // A: row within 16x4 tile
    const int koff = (lane < 16) ? 0 : 2;     // A/B: K=0,1 (lo lanes) / K=2,3 (hi)

    // ---- load M_r (256 KB) into LDS ----------------------------------------
#if __has_builtin(__builtin_amdgcn_tensor_load_to_lds) && __has_builtin(__builtin_amdgcn_s_wait_tensorcnt)
    if (wave == 0) {
        // Tensor DMA: 256x256 fp32 tile, LDS pad 1 DWORD per 256 DWORDs
        // => LDS rows at stride 257 (matches MS_STRIDE).
        unsigned long long ga = (unsigned long long)(size_t)(tm + (size_t)r * (EMB * EMB));
        unsigned ldsa = (unsigned)(size_t)(void*)Ms;   // LDS byte offset (low 32 bits)
        v4u g0;
        g0[0] = 1u;                                       // count=1, user descriptor
        g0[1] = ldsa;                                     // lds_addr
        g0[2] = (unsigned)(ga & 0xFFFFFFFFu);             // global_addr[31:0]
        g0[3] = (unsigned)((ga >> 32) & 0x01FFFFFFu)      // global_addr[56:32]
              | (2u << 30);                               // type=2 (image)
        v8i g1;
        g1[0] = (int)((2u << 16)      // data_size = 4B
                    | (1u << 20)      // pad_enable
                    | (7u << 22)      // pad_interval code 7 = 256 DWORDs
                    | (0u << 25));    // pad_amount  code 0 = 1 DWORD
        g1[1] = (int)(256u << 16);    // tensor_dim0 = 256 (bits 63:48)
        g1[2] = (int)(256u << 16);    // tensor_dim1 = 256 (bits 95:80)
        g1[3] = (int)(256u << 16);    // tile_dim0   = 256 (bits 127:112)
        g1[4] = 256;                  // tile_dim1   = 256 (bits 143:128)
        g1[5] = 256;                  // tensor_dim0_stride = 256 (bits 207:160)
        g1[6] = 0;
        g1[7] = 0;
        v4i gz4 = {0, 0, 0, 0};               // groups 2/3 unused (2D tensor)
        v8i gz8 = {0, 0, 0, 0, 0, 0, 0, 0};   // extra group (clang-23 6-arg form)
        __builtin_amdgcn_tensor_load_to_lds(g0, g1, gz4, gz4, gz8, 0);
        __builtin_amdgcn_s_wait_tensorcnt(0);
    }
    __syncthreads();
#else
    {
        const float4* Mg4 = (const float4*)(tm + (size_t)r * (EMB * EMB));
        for (int idx = tid; idx < (EMB * EMB) / 4; idx += 256) {
            float4 v = Mg4[idx];
            int row = idx >> 6;
            float* dst = &Ms[row * MS_STRIDE + ((idx & 63) << 2)];
            dst[0] = v.x; dst[1] = v.y; dst[2] = v.z; dst[3] = v.w;
        }
        __syncthreads();
    }
#endif

    const int nchunks = (nr + 15) >> 4;

    // =====================  Phase 1:  T = X2 @ M - X1  =======================
    for (int ch = 0; ch < nchunks; ++ch) {
        const int rows = (nr - ch * 16 < 16) ? (nr - ch * 16) : 16;

        for (int idx = tid; idx < 16 * (EMB / 4); idx += 256) {
            int rw = idx >> 6, c4 = idx & 63;
            float4 v = make_float4(0.f, 0.f, 0.f, 0.f);
            if (rw < rows) {
                int gb = perm[base + ch * 16 + rw];
                v = ((const float4*)x2)[(size_t)gb * (EMB / 4) + c4];
            }
            ((float4*)&Xs[rw * XS_STRIDE])[c4] = v;
        }
        __syncthreads();

        for (int t = 0; t < 2; ++t) {                  // 8 waves x 2 = 16 N-tiles
            const int nt   = wave * 2 + t;
            const int ncol = nt * 16 + lcol;

            v8f c;                                     // C := -x1 tile
            for (int v = 0; v < 8; ++v) {
                int rw = v + lrowbase;
                float xv = 0.f;
                if (rw < rows) {
                    int gb = perm[base + ch * 16 + rw];
                    xv = x1[(size_t)gb * EMB + ncol];
                }
                c[v] = -xv;
            }
            for (int k4 = 0; k4 < EMB / 4; ++k4) {
                v2f a, b;
                a.x = Xs[arow * XS_STRIDE + k4 * 4 + koff];
                a.y = Xs[arow * XS_STRIDE + k4 * 4 + koff + 1];
                b.x = Ms[(k4 * 4 + koff)     * MS_STRIDE + ncol];
                b.y = Ms[(k4 * 4 + koff + 1) * MS_STRIDE + ncol];
                c = __builtin_amdgcn_wmma_f32_16x16x4_f32(
                        false, a, false, b, (short)0, c, false, false);
            }
            for (int v = 0; v < 8; ++v) {
                int rw = v + lrowbase;
                if (rw < rows)
                    T[(size_t)(base + ch * 16 + rw) * EMB + ncol] = c[v];
            }
        }
        __syncthreads();
    }

    __threadfence();   // make T visible for the phase-3 reload
    __syncthreads();

    // ==========  Phase 2: blocked in-place Gauss-Jordan inverse  ============
    // Panel of 16 columns factored scalar-style (identity-substitution trick:
    // panel ends up holding U = E*[e_k0..e_k0+15]).  Trailing columns updated
    // with rank-16 WMMA GEMM:  A[:, j not in panel] += (U - S) * R,
    // where R = original pivot rows (snapshotted into Xs).
    const int pcol  = tid & 15;
    const int prow0 = tid >> 4;
    for (int k0 = 0; k0 < EMB; k0 += 16) {
        const int pt = k0 >> 4;

        // ---- unblocked GJ on panel columns [k0, k0+16) ----
        for (int k = 0; k < 16; ++k) {
            const int kg = k0 + k;
            __syncthreads();                       // prior writes visible
            float p = Ms[kg * MS_STRIDE + kg];
            fcol[tid] = Ms[tid * MS_STRIDE + kg];  // conflict-free column read
            float pinv = 1.0f / p;
            __syncthreads();
            if (tid < 16) {
                float v = (tid == k) ? pinv
                                     : Ms[kg * MS_STRIDE + k0 + tid] * pinv;
                prow[tid] = v;
                Ms[kg * MS_STRIDE + k0 + tid] = v;
            }
            __syncthreads();
            float pv = prow[pcol];
            for (int s = 0; s < 16; ++s) {
                int i = prow0 + (s << 4);
                if (i == kg) continue;
                float f   = fcol[i];               // LDS broadcast
                float val = (pcol == k) ? 0.f : Ms[i * MS_STRIDE + k0 + pcol];
                Ms[i * MS_STRIDE + k0 + pcol] = fmaf(-f, pv, val);
            }
        }
        __syncthreads();

        // ---- snapshot pivot rows R into Xs (trailing cols still original) ----
        for (int idx = tid; idx < 16 * (EMB / 4); idx += 256) {
            int rw = idx >> 6, c4 = (idx & 63) << 2;
            const float* src = &Ms[(k0 + rw) * MS_STRIDE + c4];
            float*       dst = &Xs[rw * XS_STRIDE + c4];
            dst[0] = src[0]; dst[1] = src[1]; dst[2] = src[2]; dst[3] = src[3];
        }
        __syncthreads();

        // ---- rank-16 trailing update on WMMA: 240 tiles, 30 per wave ----
        for (int tt = 0; tt < 30; ++tt) {
            int tile = wave * 30 + tt;
            int rb   = tile / 15;                  // row block 0..15
            int jt2  = tile % 15;
            int jt   = jt2 + (jt2 >= pt ? 1 : 0);  // skip panel column-tile
            int i0   = rb * 16;
            int ncol = jt * 16 + lcol;

            v8f c;                                 // C := existing tile
            for (int v = 0; v < 8; ++v)
                c[v] = Ms[(i0 + v + lrowbase) * MS_STRIDE + ncol];

            for (int k4 = 0; k4 < 4; ++k4) {
                int kk = k4 * 4 + koff;
                int ia = i0 + arow;
                v2f a, b;
                a.x = Ms[ia * MS_STRIDE + k0 + kk]     - ((ia == k0 + kk)     ? 1.f : 0.f);
                a.y = Ms[ia * MS_STRIDE + k0 + kk + 1] - ((ia == k0 + kk + 1) ? 1.f : 0.f);
                b.x = Xs[kk       * XS_STRIDE + ncol];
                b.y = Xs[(kk + 1) * XS_STRIDE + ncol];
                c = __builtin_amdgcn_wmma_f32_16x16x4_f32(
                        false, a, false, b, (short)0, c, false, false);
            }
            for (int v = 0; v < 8; ++v)
                Ms[(i0 + v + lrowbase) * MS_STRIDE + ncol] = c[v];
        }
        __syncthreads();
    }

    // =====================  Phase 3:  OUT = T @ Minv  ========================
    for (int ch = 0; ch < nchunks; ++ch) {
        const int rows = (nr - ch * 16 < 16) ? (nr - ch * 16) : 16;

        for (int idx = tid; idx < 16 * (EMB / 4); idx += 256) {
            int rw = idx >> 6, c4 = idx & 63;
            float4 v = make_float4(0.f, 0.f, 0.f, 0.f);
            if (rw < rows)
                v = ((const float4*)T)[(size_t)(base + ch * 16 + rw) * (EMB / 4) + c4];
            ((float4*)&Xs[rw * XS_STRIDE])[c4] = v;
        }
        __syncthreads();

        for (int t = 0; t < 2; ++t) {
            const int nt   = wave * 2 + t;
            const int ncol = nt * 16 + lcol;

            v8f c;
            for (int v = 0; v < 8; ++v) c[v] = 0.f;

            for (int k4 = 0; k4 < EMB / 4; ++k4) {
                v2f a, b;
                a.x = Xs[arow * XS_STRIDE + k4 * 4 + koff];
                a.y = Xs[arow * XS_STRIDE + k4 * 4 + koff + 1];
                b.x = Ms[(k4 * 4 + koff)     * MS_STRIDE + ncol];
                b.y = Ms[(k4 * 4 + koff + 1) * MS_STRIDE + ncol];
                c = __builtin_amdgcn_wmma_f32_16x16x4_f32(
                        false, a, false, b, (short)0, c, false, false);
            }
            for (int v = 0; v < 8; ++v) {
                int rw = v + lrowbase;
                if (rw < rows) {
                    int gb = perm[base + ch * 16 + rw];
                    out[(size_t)gb * EMB + ncol] = c[v];
                }
            }
        }
        __syncthreads();
    }
}

// ---------------------------------------------------------------------------
extern "C" void kernel_launch(void* const* d_in, const int* in_sizes, int n_in,
                              void* d_out, int out_size, void* d_ws, size_t ws_size,
                              hipStream_t stream) {
    (void)in_sizes; (void)n_in; (void)out_size; (void)ws_size;
    // inputs: x0 (unused in mode 0), x1, x2, rel (int32), transfer_matrix
    const float* x1  = (const float*)d_in[1];
    const float* x2  = (const float*)d_in[2];
    const int*   rel = (const int*)d_in[3];
    const float* tm  = (const float*)d_in[4];
    float*       out = (float*)d_out;

    int*   wsi     = (int*)d_ws;
    int*   counts  = wsi + WS_COUNTS;
    int*   offsets = wsi + WS_OFFSETS;
    int*   cursor  = wsi + WS_CURSOR;
    int*   perm    = wsi + WS_PERM;
    float* T       = (float*)d_ws + WS_T;

    const size_t ldsBytes =
        (size_t)(EMB * MS_STRIDE + 16 * XS_STRIDE + EMB + 16) * sizeof(float); // ~281 KB

    // opt in to >64KB dynamic LDS (CDNA5: up to 320 KB per workgroup); idempotent
    (void)hipFuncSetAttribute((const void*)k_transr,
                              hipFuncAttributeMaxDynamicSharedMemorySize,
                              (int)ldsBytes);

    k_zero   <<<(RELS + 255) / 256, 256, 0, stream>>>(counts);
    k_count  <<<BATCH / 256,        256, 0, stream>>>(rel, counts);
    k_scan   <<<1,                    1, 0, stream>>>(counts, offsets, cursor);
    k_scatter<<<BATCH / 256,        256, 0, stream>>>(rel, cursor, perm);
    k_transr <<<RELS, 256, ldsBytes, stream>>>(x1, x2, tm, offsets, perm, T, out);
}